// WeightedSAGEConv_59846074302804
// MI455X (gfx1250) — compile-verified
//
#include <hip/hip_runtime.h>
#include <hip/hip_bf16.h>

typedef __attribute__((ext_vector_type(2))) float v2f;
typedef __attribute__((ext_vector_type(8))) float v8f;

#define DIN  128
#define DOUT 128
#define DCAT 256

// ---------------------------------------------------------------------------
// Kernel 0: zero the aggregation scratch (agg + ws are contiguous in d_ws).
// n elements is a multiple of 4 here (n_dst*128 + n_dst, n_dst a multiple of 4).
// ---------------------------------------------------------------------------
__global__ void zero_ws_kernel(float4* __restrict__ p, long long n4) {
  long long i = (long long)blockIdx.x * blockDim.x + threadIdx.x;
  long long stride = (long long)gridDim.x * blockDim.x;
  const float4 z = make_float4(0.f, 0.f, 0.f, 0.f);
  for (; i < n4; i += stride) p[i] = z;
}

// ---------------------------------------------------------------------------
// Kernel 1: n = relu(h_src @ Q_w^T + Q_b)  [N_src x 128]
// Block = 256 threads = 8 waves. Wave w owns output columns [16w, 16w+16).
// It preloads ALL 32 K-step B fragments of Q_w for its column block into
// registers (64 VGPRs) and then streams row tiles with V_WMMA_F32_16X16X4_F32.
//
// f32 WMMA fragment layout (wave32), 16x16x4:
//   A (16x4):  lane half h=lane>>4, ml=lane&15.  VGPR0 = A[ml][k0+2h],
//              VGPR1 = A[ml][k0+2h+1]  -> contiguous float2 per lane.
//   B (4x16):  VGPR0 = B[k0+2h][n0+ml], VGPR1 = B[k0+2h+1][n0+ml];
//              B[k][n] = Q_w[n][k] (row-major) -> contiguous float2 per lane.
//   C/D: VGPR r -> row m0+r (lanes 0-15) / m0+8+r (lanes 16-31), col n0+ml.
// N_src = 100000 is divisible by 16, so no row tail handling is needed.
// ---------------------------------------------------------------------------
__global__ __launch_bounds__(256)
void q_gemm_relu(const float* __restrict__ hsrc, const float* __restrict__ Qw,
                 const float* __restrict__ Qb, float* __restrict__ nout,
                 int ntiles) {
  const int lane = threadIdx.x & 31;
  const int wave = threadIdx.x >> 5;   // 0..7 -> N block
  const int half = lane >> 4;
  const int ml   = lane & 15;
  const int n0   = wave * 16;

  v2f bf[32];
#pragma unroll
  for (int k = 0; k < 32; ++k)
    bf[k] = *(const v2f*)(Qw + (size_t)(n0 + ml) * DIN + 4 * k + 2 * half);

  const float bias = Qb[n0 + ml];

  for (int t = blockIdx.x; t < ntiles; t += gridDim.x) {
    const int m0 = t * 16;
    const float* arow = hsrc + (size_t)(m0 + ml) * DIN + 2 * half;
    v8f acc = {};
#pragma unroll
    for (int k = 0; k < 32; ++k) {
      v2f a = *(const v2f*)(arow + 4 * k);
      acc = __builtin_amdgcn_wmma_f32_16x16x4_f32(
          /*neg_a=*/false, a, /*neg_b=*/false, bf[k],
          /*c_mod=*/(short)0, acc, /*reuse_a=*/false, /*reuse_b=*/false);
    }
#pragma unroll
    for (int r = 0; r < 8; ++r) {
      const int row = m0 + r + half * 8;
      nout[(size_t)row * DOUT + n0 + ml] = fmaxf(acc[r] + bias, 0.0f);
    }
  }
}

// ---------------------------------------------------------------------------
// Kernel 2: edge gather / weighted scatter-add.
//   agg[dst[e]][d] += n[src[e]][d] * w[e];  ws[dst[e]] += w[e]
// One WAVE per edge: 32 lanes x float4 = 128 dims. The gather stays a fully
// coalesced 512B read per edge; src/dst/w become wave-uniform scalars (loaded
// once instead of 128x). agg (51 MB) is L2-resident -> f32 atomics ride L2.
// ---------------------------------------------------------------------------
__global__ __launch_bounds__(256)
void edge_scatter(const float* __restrict__ nsrc, const float* __restrict__ w,
                  const int* __restrict__ src, const int* __restrict__ dst,
                  float* __restrict__ agg, float* __restrict__ ws, int E) {
  const int lane = threadIdx.x & 31;
  const long long e = (long long)blockIdx.x * 8 + (threadIdx.x >> 5);
  if (e >= E) return;
  const int s = src[e];
  const int t = dst[e];
  const float we = w[e];
  const float4 v = *(const float4*)(nsrc + (size_t)s * DOUT + lane * 4);
  float* dp = agg + (size_t)t * DOUT + lane * 4;
  atomicAdd(dp + 0, v.x * we);
  atomicAdd(dp + 1, v.y * we);
  atomicAdd(dp + 2, v.z * we);
  atomicAdd(dp + 3, v.w * we);
  if (lane == 0) atomicAdd(ws + t, we);
}

// ---------------------------------------------------------------------------
// Kernel 3: z = relu(concat([agg/max(ws,1), h_dst]) @ W_w^T + W_b); z /= ||z||
// Whole W_w (128x256 f32 = 128 KB) is staged into LDS once per block (WGP has
// 320 KB LDS) and shared by 8 waves. Each wave computes a full 16x128 output
// row block (8 accumulators, K=256 -> 64 WMMA steps; the A fragment is reused
// across the 8 N blocks). The agg/ws division is fused into the A load.
// Row norms: per-lane sum of squares over its 8 column blocks, then
// __shfl_xor(1,2,4,8) reduces across the 16 lanes of each half-wave (rows r
// live in lanes 0-15, rows r+8 in lanes 16-31, so halves never mix rows).
// ---------------------------------------------------------------------------
__global__ __launch_bounds__(256)
void w_gemm_norm(const float* __restrict__ agg, const float* __restrict__ ws,
                 const float* __restrict__ hdst, const float* __restrict__ Ww,
                 const float* __restrict__ Wb, float* __restrict__ out,
                 int ntiles) {
  __shared__ float wlds[DOUT * DCAT];  // 128 KB: full W_w
  {
    const float4* s = (const float4*)Ww;
    float4* d4 = (float4*)wlds;
    for (int i = threadIdx.x; i < DOUT * DCAT / 4; i += blockDim.x) d4[i] = s[i];
  }
  __syncthreads();

  const int lane = threadIdx.x & 31;
  const int wave = threadIdx.x >> 5;
  const int half = lane >> 4;
  const int ml   = lane & 15;

  for (int t = blockIdx.x * 8 + wave; t < ntiles; t += gridDim.x * 8) {
    const int m0  = t * 16;
    const int row = m0 + ml;                       // A-matrix row for this lane
    const float invw = 1.0f / fmaxf(ws[row], 1.0f);
    const float* aggrow = agg  + (size_t)row * DOUT + 2 * half;
    const float* hrow   = hdst + (size_t)row * DIN  + 2 * half;

    v8f acc[8] = {};
    for (int k = 0; k < 64; ++k) {                 // K = 256, 4 per step
      const int k0 = 4 * k;
      v2f a;
      if (k0 < DOUT) {                             // uniform branch per wave
        a = *(const v2f*)(aggrow + k0);
        a[0] *= invw;
        a[1] *= invw;
      } else {
        a = *(const v2f*)(hrow + (k0 - DOUT));
      }
#pragma unroll
      for (int nb = 0; nb < 8; ++nb) {
        v2f b = *(const v2f*)(wlds + (size_t)(nb * 16 + ml) * DCAT + k0 + 2 * half);
        acc[nb] = __builtin_amdgcn_wmma_f32_16x16x4_f32(
            false, a, false, b, (short)0, acc[nb], false, false);
      }
    }

    // Bias + ReLU + per-row sum of squares (8 column blocks per lane).
    float sums[8];
#pragma unroll
    for (int r = 0; r < 8; ++r) sums[r] = 0.0f;
#pragma unroll
    for (int nb = 0; nb < 8; ++nb) {
      const float bias = Wb[nb * 16 + ml];
#pragma unroll
      for (int r = 0; r < 8; ++r) {
        const float v = fmaxf(acc[nb][r] + bias, 0.0f);
        acc[nb][r] = v;
        sums[r] += v * v;
      }
    }
    // Reduce across 16 lanes of each half-wave; convert to 1/||z|| (0 -> 1).
#pragma unroll
    for (int r = 0; r < 8; ++r) {
      float s = sums[r];
      s += __shfl_xor(s, 1);
      s += __shfl_xor(s, 2);
      s += __shfl_xor(s, 4);
      s += __shfl_xor(s, 8);
      sums[r] = (s > 0.0f) ? (1.0f / sqrtf(s)) : 1.0f;
    }
#pragma unroll
    for (int nb = 0; nb < 8; ++nb) {
#pragma unroll
      for (int r = 0; r < 8; ++r) {
        const int orow = m0 + r + half * 8;
        out[(size_t)orow * DOUT + nb * 16 + ml] = acc[nb][r] * sums[r];
      }
    }
  }
}

// ---------------------------------------------------------------------------
// Launch
// ---------------------------------------------------------------------------
extern "C" void kernel_launch(void* const* d_in, const int* in_sizes, int n_in,
                              void* d_out, int out_size, void* d_ws, size_t ws_size,
                              hipStream_t stream) {
  const float* h_src   = (const float*)d_in[0];
  const float* h_dst   = (const float*)d_in[1];
  const float* weights = (const float*)d_in[2];
  const float* Q_w     = (const float*)d_in[3];
  const float* Q_b     = (const float*)d_in[4];
  const float* W_w     = (const float*)d_in[5];
  const float* W_b     = (const float*)d_in[6];
  const int*   src_idx = (const int*)d_in[7];
  const int*   dst_idx = (const int*)d_in[8];

  const int n_src = in_sizes[0] / DIN;
  const int n_dst = in_sizes[1] / DIN;
  const int E     = in_sizes[2];

  // Workspace layout: n [n_src*128] | agg [n_dst*128] | ws [n_dst]
  float* nbuf = (float*)d_ws;
  float* agg  = nbuf + (size_t)n_src * DOUT;
  float* wsum = agg  + (size_t)n_dst * DOUT;

  const long long zelems = (long long)n_dst * DOUT + n_dst;  // agg+ws contiguous
  zero_ws_kernel<<<2048, 256, 0, stream>>>((float4*)agg, zelems / 4);

  q_gemm_relu<<<1024, 256, 0, stream>>>(h_src, Q_w, Q_b, nbuf, n_src / 16);

  edge_scatter<<<(E + 7) / 8, 256, 0, stream>>>(
      nbuf, weights, src_idx, dst_idx, agg, wsum, E);

  w_gemm_norm<<<256, 256, 0, stream>>>(agg, wsum, h_dst, W_w, W_b,
                                       (float*)d_out, n_dst / 16);
}